// Conduits_77876347011668
// MI455X (gfx1250) — compile-verified
//
#include <hip/hip_runtime.h>
#include <math.h>

#ifndef M_PI
#define M_PI 3.14159265358979323846
#endif

// Match the JAX reference op-for-op: it performs mul and add as separately
// rounded f32 ops, so forbid FMA contraction (memory-bound kernel; FMA buys
// no performance here anyway).
#pragma clang fp contract(off)

#define TB 256                 // 8 wave32 waves per block on CDNA5
#define NZF 1.0e-12f           // NONZERO

// ---------------------------------------------------------------------------
// Node setup: overburden, eff = ovb - p0, zero gradient accumulator
// ---------------------------------------------------------------------------
__global__ void k_node_init(const float* __restrict__ ice, const float* __restrict__ p0,
                            float* __restrict__ ovb, float* __restrict__ eff,
                            float* __restrict__ g, int N)
{
    int n = blockIdx.x * blockDim.x + threadIdx.x;
    if (n >= N) return;
    float o = 8995.77f * ice[n];          // RHO_I * G (rounded like reference)
    ovb[n] = o;
    eff[n] = o - p0[n];
    g[n]   = 0.0f;
}

// ---------------------------------------------------------------------------
// Link setup: base_gradient (masked by status)
// ---------------------------------------------------------------------------
__global__ void k_link_init(const float* __restrict__ ovb, const float* __restrict__ bed,
                            const float* __restrict__ len,
                            const int* __restrict__ head, const int* __restrict__ tail,
                            const int* __restrict__ status,
                            float* __restrict__ bg, int L)
{
    int l = blockIdx.x * blockDim.x + threadIdx.x;
    if (l >= L) return;
    float b = 0.0f;
    if (status[l] == 0) {
        int h = head[l], t = tail[l];
        float Ln = len[l];
        b = -((ovb[h] - ovb[t]) / Ln) - 9810.0f * ((bed[h] - bed[t]) / Ln);
    }
    bg[l] = b;
}

// ---------------------------------------------------------------------------
// RK4 conduit evolution. Pressure is fixed at p0 across all 4 stages, so
// hg0 / conduit_p / rs are stage-invariant.  Arithmetic ordered exactly as
// the reference: q = ((FLOW_C*S^1.25)*rs)*hg ; melt = (MELT_C*q)*hg ;
// creep = (CLOSURE_C*cp^3)*S ; rate = (melt + gap) - creep.
// Must run while eff[] still equals ovb - p0.
// ---------------------------------------------------------------------------
__global__ void k_rk4(const float* __restrict__ eff, const float* __restrict__ bg,
                      const float* __restrict__ len,
                      const float* __restrict__ S0a, const float* __restrict__ vel,
                      const int* __restrict__ head, const int* __restrict__ tail,
                      const int* __restrict__ status, const int* __restrict__ dtp,
                      float flow_c, float melt_c, float closure_c,
                      float* __restrict__ outC, int L)
{
    int l = blockIdx.x * blockDim.x + threadIdx.x;
    if (l >= L) return;
    if (status[l] != 0) { outC[l] = 0.0f; return; }   // inactive -> 0

    float dt = (float)(*dtp);
    int h = head[l], t = tail[l];
    float eh = eff[h], et = eff[t];
    float Ln = len[l];
    float hg = bg[l] + (eh - et) / Ln;
    float cp = 0.5f * (eh + et);
    float ab  = fabsf(hg);
    float anz = fmaxf(ab, NZF);                       // |nzp|
    float rs  = 1.0f / sqrtf(anz);                    // |nzp|^-0.5
    float gap = vel[l] * 0.1f;                        // STEP_H
    float cc  = closure_c * (cp * cp * cp);           // GLENS_N = 3 (integer power)
    float S   = S0a[l];

    #define RATE(Sv) ((melt_c * ((flow_c * powf((Sv), 1.25f)) * rs * (Sv >= 0.f ? hg : hg)) * hg \
                       + gap) - cc * (Sv))
    // NOTE: powf(Sv,1.25) yields NaN for Sv<0, matching jnp.power semantics.
    float k1 = (melt_c * ((flow_c * powf(S, 1.25f)) * rs * hg) * hg + gap) - cc * S;
    float s2 = S + k1 * dt / 2.0f;
    float k2 = (melt_c * ((flow_c * powf(s2, 1.25f)) * rs * hg) * hg + gap) - cc * s2;
    float s3 = S + k2 * dt / 2.0f;
    float k3 = (melt_c * ((flow_c * powf(s3, 1.25f)) * rs * hg) * hg + gap) - cc * s3;
    float s4 = S + k3 * dt;
    float k4 = (melt_c * ((flow_c * powf(s4, 1.25f)) * rs * hg) * hg + gap) - cc * s4;
    #undef RATE

    float ksum = k1 + 2.0f * k2 + 2.0f * k3 + k4;     // left-assoc like reference
    float Sn = S + (dt * ksum) / 6.0f;
    outC[l] = (Sn < 0.0f) ? 0.0f : Sn;                // ternary: keep NaN like jnp.where
}

// ---------------------------------------------------------------------------
// Per-iteration link pass: discharge q, adjoint weight w = dq/dhg / len,
// and zero the link-gradient accumulator for the upcoming scatter.
// ---------------------------------------------------------------------------
__global__ void k_link_q(const float* __restrict__ eff, const float* __restrict__ bg,
                         const float* __restrict__ len, const float* __restrict__ S0a,
                         const int* __restrict__ head, const int* __restrict__ tail,
                         float flow_c,
                         float* __restrict__ q, float* __restrict__ w,
                         float* __restrict__ gbar, int L)
{
    int l = blockIdx.x * blockDim.x + threadIdx.x;
    if (l >= L) return;
    float Ln = len[l];
    float hg = bg[l] + (eff[head[l]] - eff[tail[l]]) / Ln;
    float c  = flow_c * powf(S0a[l], 1.25f);          // FLOW_C * S^1.25
    float ab = fabsf(hg);
    float anz = fmaxf(ab, NZF);
    float rs  = 1.0f / sqrtf(anz);                    // |nzp|^-0.5
    q[l] = c * rs * hg;
    // d q / d hg : inside the clamp nzp is constant -> c*rs ; else c*0.5*rs
    float dq = (ab < NZF) ? (c * rs) : (c * 0.5f * rs);
    w[l] = dq / Ln;
    gbar[l] = 0.0f;
}

// ---------------------------------------------------------------------------
// Node flux + adjoint scatter to links. Grid-stride loop so that
// global_prefetch_b8 (gfx1250) can pull the next index rows into L0/L2
// while this iteration's random q-gathers and f32 atomics are in flight.
// ---------------------------------------------------------------------------
__global__ void k_node_flux(const int* __restrict__ lat, const int* __restrict__ dirs,
                            const float* __restrict__ melt, const float* __restrict__ q,
                            float* __restrict__ gbar, float scale /* LR/N */, int N)
{
    int n0 = blockIdx.x * blockDim.x + threadIdx.x;
    int stride = gridDim.x * blockDim.x;
    for (int n = n0; n < N; n += stride) {
        int nxt = n + stride;
        if (nxt < N) {
            __builtin_prefetch(lat  + 6 * (size_t)nxt, 0, 3);   // global_prefetch_b8
            __builtin_prefetch(dirs + 6 * (size_t)nxt, 0, 3);
            __builtin_prefetch(melt + nxt, 0, 3);
        }
        const int2* lr = (const int2*)(lat  + 6 * (size_t)n);   // 8B-aligned rows
        const int2* dr = (const int2*)(dirs + 6 * (size_t)n);
        int2 l0 = lr[0], l1 = lr[1], l2 = lr[2];
        int2 d0 = dr[0], d1 = dr[1], d2 = dr[2];
        float f0 = (float)d0.x, f1 = (float)d0.y, f2 = (float)d1.x;
        float f3 = (float)d1.y, f4 = (float)d2.x, f5 = (float)d2.y;
        float nf = f0 * q[l0.x] + f1 * q[l0.y] + f2 * q[l1.x]
                 + f3 * q[l1.y] + f4 * q[l2.x] + f5 * q[l2.y] - melt[n];
        float t = tanhf(nf) * scale;                  // d(logcosh)/dnf * LR/N
        if (d0.x) atomicAdd(gbar + l0.x, f0 * t);
        if (d0.y) atomicAdd(gbar + l0.y, f1 * t);
        if (d1.x) atomicAdd(gbar + l1.x, f2 * t);
        if (d1.y) atomicAdd(gbar + l1.y, f3 * t);
        if (d2.x) atomicAdd(gbar + l2.x, f4 * t);
        if (d2.y) atomicAdd(gbar + l2.y, f5 * t);
    }
}

// ---------------------------------------------------------------------------
// Link adjoint -> node gradient scatter.  dhg/dp_head = -1/len, dp_tail = +1/len
// (1/len already folded into w).
// ---------------------------------------------------------------------------
__global__ void k_link_scatter(const float* __restrict__ gbar, const float* __restrict__ w,
                               const int* __restrict__ head, const int* __restrict__ tail,
                               float* __restrict__ g, int L)
{
    int l = blockIdx.x * blockDim.x + threadIdx.x;
    if (l >= L) return;
    float lam = gbar[l] * w[l];
    if (lam != 0.0f) {
        atomicAdd(g + head[l], -lam);
        atomicAdd(g + tail[l],  lam);
    }
}

// ---------------------------------------------------------------------------
// p <- p - g  is eff <- eff + g (eff = ovb - p); zero g for next iteration.
// ---------------------------------------------------------------------------
__global__ void k_node_update(float* __restrict__ eff, float* __restrict__ g, int N)
{
    int n = blockIdx.x * blockDim.x + threadIdx.x;
    if (n >= N) return;
    eff[n] += g[n];
    g[n] = 0.0f;
}

// ---------------------------------------------------------------------------
// Final pressure: p = ovb - eff, clamped at 0 (NaN-preserving ternary).
// ---------------------------------------------------------------------------
__global__ void k_final(const float* __restrict__ ovb, const float* __restrict__ eff,
                        float* __restrict__ outP, int N)
{
    int n = blockIdx.x * blockDim.x + threadIdx.x;
    if (n >= N) return;
    float p = ovb[n] - eff[n];
    outP[n] = (p < 0.0f) ? 0.0f : p;
}

// ---------------------------------------------------------------------------
extern "C" void kernel_launch(void* const* d_in, const int* in_sizes, int n_in,
                              void* d_out, int out_size, void* d_ws, size_t ws_size,
                              hipStream_t stream)
{
    const float* ice    = (const float*)d_in[0];
    const float* bed    = (const float*)d_in[1];
    const float* melt   = (const float*)d_in[2];
    const float* vel    = (const float*)d_in[3];
    const float* p0     = (const float*)d_in[4];
    const float* S0a    = (const float*)d_in[5];
    const float* len    = (const float*)d_in[6];
    const int*   head   = (const int*)d_in[7];
    const int*   tail   = (const int*)d_in[8];
    const int*   lat    = (const int*)d_in[9];
    const int*   dirs   = (const int*)d_in[10];
    const int*   status = (const int*)d_in[11];
    const int*   dtp    = (const int*)d_in[12];

    const int N = in_sizes[0];
    const int L = in_sizes[3];

    // Exact constants computed in double on host, rounded once to f32
    const float flow_c    = (float)(pow(2.0, 0.25) * sqrt(M_PI + 2.0) /
                                    (pow(M_PI, 0.25) * sqrt(1000.0 * 0.0375)));
    const float melt_c    = (float)(1.0 / (917.0 * 335000.0));
    const float closure_c = (float)(2.0 * 6.0e-24 / 27.0);
    const float scale     = (float)(1.0e3 / (double)N);   // SOLVER_LR / N

    // Workspace layout: 3N + 4L floats (= 60 MB at N=1e6, L=3e6)
    float* ws   = (float*)d_ws;
    float* ovb  = ws;          // [N]
    float* eff  = ovb + N;     // [N]  (mutated by solver)
    float* g    = eff + N;     // [N]  node gradient accumulator
    float* bg   = g + N;       // [L]  base gradient
    float* q    = bg + L;      // [L]  discharge
    float* w    = q + L;       // [L]  dq/dhg / len
    float* gbar = w + L;       // [L]  dL/dq accumulator

    float* outP = (float*)d_out;   // [N] new pressure
    float* outC = outP + N;        // [L] new conduits

    const dim3 tb(TB);
    const int gn = (N + TB - 1) / TB;
    const int gl = (L + TB - 1) / TB;
    const int gn4 = (N + TB * 4 - 1) / (TB * 4);   // grid-stride: 4 nodes/thread

    k_node_init<<<gn, tb, 0, stream>>>(ice, p0, ovb, eff, g, N);
    k_link_init<<<gl, tb, 0, stream>>>(ovb, bed, len, head, tail, status, bg, L);

    // RK4 before the solver mutates eff (it needs eff0 = ovb - p0)
    k_rk4<<<gl, tb, 0, stream>>>(eff, bg, len, S0a, vel, head, tail, status, dtp,
                                 flow_c, melt_c, closure_c, outC, L);

    for (int it = 0; it < 10; ++it) {
        k_link_q<<<gl, tb, 0, stream>>>(eff, bg, len, S0a, head, tail, flow_c,
                                        q, w, gbar, L);
        k_node_flux<<<gn4, tb, 0, stream>>>(lat, dirs, melt, q, gbar, scale, N);
        k_link_scatter<<<gl, tb, 0, stream>>>(gbar, w, head, tail, g, L);
        k_node_update<<<gn, tb, 0, stream>>>(eff, g, N);
    }

    k_final<<<gn, tb, 0, stream>>>(ovb, eff, outP, N);
}